// SequentialSSM_73289321939298
// MI455X (gfx1250) — compile-verified
//
#include <hip/hip_runtime.h>

// Sequential SSM scan (Mamba-style), FP32.
// Roofline: deltaA+deltaB = 310 MB, total ~330 MB @ 23.3 TB/s => ~14 us floor;
// only ~194 MFLOP => purely HBM-bound. WMMA (16x16 shapes) cannot accelerate a
// diagonal recurrence; the MI455X features that matter here are data movement:
//   - global_load_async_to_lds_b32 (ASYNCcnt) for the C transpose staging
//   - tensor_load_to_lds (TDM, TENSORcnt) for the contiguous x tile staging
//   - global_prefetch_b8 deep stream prefetch for the dA/dB streams
// Mapping: 4 lanes per (b,d) channel, 4 of 16 states per lane in a float4
//   -> 49152 lanes = 1536 wave32 waves, contiguous 64B/group/step streams.

#define BATCH   32
#define DINNER  384
#define LSEQ    197
#define DSTATE  16

#define THREADS 128
#define CH_PER_BLOCK (THREADS / 4)               // 32 channels per block
#define BLOCKS_PER_BATCH (DINNER / CH_PER_BLOCK) // 12
#define XTILE (CH_PER_BLOCK * LSEQ)              // 6304 floats, contiguous in memory
#define PF_DIST 16                               // prefetch distance (1 KB / stream)

#if defined(__has_builtin)
#if __has_builtin(__builtin_amdgcn_global_load_async_to_lds_b32)
#define HAVE_ASYNC_LDS 1
#endif
#if __has_builtin(__builtin_amdgcn_tensor_load_to_lds) && __has_builtin(__builtin_amdgcn_s_wait_tensorcnt)
#define HAVE_TDM 1
#endif
#endif

typedef unsigned int tdm_v4u __attribute__((ext_vector_type(4)));
typedef int          tdm_v8i __attribute__((ext_vector_type(8)));
typedef int          tdm_v4i __attribute__((ext_vector_type(4)));

__global__ __launch_bounds__(THREADS)
void ssm_scan_kernel(const float* __restrict__ x,
                     const float* __restrict__ dA,
                     const float* __restrict__ dB,
                     const float* __restrict__ C,
                     const float* __restrict__ D,
                     float* __restrict__ y)
{
    // ldsC[t*16+s]: stride 16 floats keeps float4 reads 16B-aligned; within a
    // time step lanes touch only 4 distinct 16B chunks -> conflict-free.
    __shared__ __align__(16) float ldsC[LSEQ * DSTATE];
    // ldsX[row*197+t]: per-wave 8 groups read banks (5*grp + t) % 64 -> distinct.
    __shared__ __align__(16) float ldsX[XTILE];

    const int b   = blockIdx.x / BLOCKS_PER_BATCH;
    const int d0  = (blockIdx.x % BLOCKS_PER_BATCH) * CH_PER_BLOCK;
    const int tid = threadIdx.x;

    // ---- Stage C[b,:,:] ([DSTATE][LSEQ] row-major) into LDS transposed [t][s].
    // Async global->LDS scatter: coalesced global reads, per-lane LDS addresses.
    {
        const float* Cb = C + (size_t)b * (DSTATE * LSEQ);
        for (int s = 0; s < DSTATE; ++s) {
            for (int t = tid; t < LSEQ; t += THREADS) {
                float* gp = const_cast<float*>(Cb + s * LSEQ + t);
                float* lp = &ldsC[t * DSTATE + s];
#ifdef HAVE_ASYNC_LDS
                __builtin_amdgcn_global_load_async_to_lds_b32(
                    (__attribute__((address_space(1))) int*)gp,
                    (__attribute__((address_space(3))) int*)lp,
                    /*imm offset*/0, /*cpol*/0);
#else
                *lp = *gp;
#endif
            }
        }
    }

    // ---- Stage the block's x tile (32 channels x 197 = one contiguous 25 KB
    // region) into LDS via the Tensor Data Mover. 1-D tile descriptor:
    //   tile_dim0 = 6304 elements of 4 B, tile_dim1 = 1 row.
#ifdef HAVE_TDM
    if (tid < 32) {  // wave 0 only (wave-uniform branch; TDM issues per-wave)
        const unsigned long long ga =
            (unsigned long long)(const void*)(x + (size_t)(b * DINNER + d0) * LSEQ);
        __attribute__((address_space(3))) float* lX3 =
            (__attribute__((address_space(3))) float*)ldsX;
        const unsigned lds_off = (unsigned)(unsigned long long)lX3;

        tdm_v4u g0;                                   // D# group 0 (128 b)
        g0[0] = 1u;                                   // count=1, user mode, no gather
        g0[1] = lds_off;                              // lds_addr (bytes)
        g0[2] = (unsigned)ga;                         // global_addr[31:0]
        g0[3] = ((unsigned)(ga >> 32) & 0x01FFFFFFu)  // global_addr[56:32]
                | (2u << 30);                         // type = 2 ("image")
        const tdm_v8i g1 = {                          // D# group 1 (256 b)
            (int)0x00020000,                   // data_size=2 (4B); mask/flags/pad = 0
            (int)((XTILE & 0xFFFF) << 16),     // tensor_dim0[15:0] in bits[63:48]
            (int)((XTILE >> 16) | (1u << 16)), // tensor_dim0[31:16] | tensor_dim1=1
            (int)(XTILE << 16),                // tile_dim0 = 6304 in bits[127:112]
            1,                                 // tile_dim1 = 1 (tile_dim2 = 0)
            XTILE,                             // tensor_dim0_stride[31:0]
            0,                                 // stride hi / tensor_dim1_stride lo
            0
        };
        const tdm_v4i gz = {0, 0, 0, 0};              // groups 2/3 unused (1-D tile)
        const tdm_v8i gz8 = {0, 0, 0, 0, 0, 0, 0, 0}; // VADDR4 slot: ISA says unused
        __builtin_amdgcn_tensor_load_to_lds(g0, g1, gz, gz, gz8, /*cpol*/0);
        __builtin_amdgcn_s_wait_tensorcnt(0);
    }
#else
    for (int i = tid; i < XTILE; i += THREADS)
        ldsX[i] = x[(size_t)(b * DINNER + d0) * LSEQ + i];
#endif

#ifdef HAVE_ASYNC_LDS
#if __has_builtin(__builtin_amdgcn_s_wait_asynccnt)
    __builtin_amdgcn_s_wait_asynccnt(0);
#else
    asm volatile("s_wait_asynccnt 0x0" ::: "memory");
#endif
#endif
    __syncthreads();

    // ---- Per-lane scan state.
    const int grp = tid >> 2;       // channel group within block (0..31)
    const int sl  = tid & 3;        // sub-lane: owns states [4*sl, 4*sl+4)
    const int d   = d0 + grp;
    const unsigned ch = (unsigned)(b * DINNER + d);

    // 32-bit byte offsets from uniform bases (arrays are ~155 MB < 2^31):
    // steers codegen to saddr + 32-bit voffset global loads.
    const unsigned offAB = (ch * (unsigned)(LSEQ * DSTATE) + (unsigned)(sl * 4)) * 4u;
    const char* dA8 = (const char*)dA;
    const char* dB8 = (const char*)dB;
    float*      py  = y + (size_t)ch * LSEQ;
    const float Dd  = D[d];
    const float* lX = &ldsX[grp * LSEQ];

    float4 h = make_float4(0.f, 0.f, 0.f, 0.f);

    // Pipeline prime: t = 0 operands.
    float4 ac = *(const float4*)(dA8 + offAB);
    float4 bc = *(const float4*)(dB8 + offAB);
    float  xc = lX[0];
    float  yreg = 0.f;

    for (int t = 0; t < LSEQ; ++t) {
        // Deep stream prefetch (gfx1250 global_prefetch_b8), clamped.
        int tp = t + PF_DIST; if (tp > LSEQ - 1) tp = LSEQ - 1;
        __builtin_prefetch(dA8 + offAB + (unsigned)tp * 64u, 0, 1);
        __builtin_prefetch(dB8 + offAB + (unsigned)tp * 64u, 0, 1);

        // Register double-buffer: issue t+1 loads before consuming t.
        const int tn = (t + 1 < LSEQ) ? (t + 1) : (LSEQ - 1);
        const float4 an = *(const float4*)(dA8 + offAB + (unsigned)tn * 64u);
        const float4 bn = *(const float4*)(dB8 + offAB + (unsigned)tn * 64u);
        const float  xn = lX[tn];

        // h = deltaB*x + deltaA*h   (4 of 16 states per lane)
        h.x = __builtin_fmaf(ac.x, h.x, bc.x * xc);
        h.y = __builtin_fmaf(ac.y, h.y, bc.y * xc);
        h.z = __builtin_fmaf(ac.z, h.z, bc.z * xc);
        h.w = __builtin_fmaf(ac.w, h.w, bc.w * xc);

        // partial dot with C_t (LDS, 16B-aligned ds_load_b128)
        const float4 c4 = *(const float4*)(&ldsC[t * DSTATE + sl * 4]);
        float p;
        p = h.x * c4.x;
        p = __builtin_fmaf(h.y, c4.y, p);
        p = __builtin_fmaf(h.z, c4.z, p);
        p = __builtin_fmaf(h.w, c4.w, p);

        // reduce across the 4 lanes of the group (wave32 cross-lane)
        p += __shfl_xor(p, 1, 32);
        p += __shfl_xor(p, 2, 32);

        const float yv = __builtin_fmaf(xc, Dd, p);   // + x*D skip term

        // Rotate ownership of y_t: every 4 steps the group stores 4 consecutive floats.
        if ((t & 3) == sl) yreg = yv;
        if ((t & 3) == 3)  py[(t - 3) + sl] = yreg;

        ac = an; bc = bn; xc = xn;
    }
    // Tail: L = 197, t = 196 has (t & 3) == 0 -> sub-lane 0 holds it.
    if (sl == 0) py[LSEQ - 1] = yreg;
}

extern "C" void kernel_launch(void* const* d_in, const int* in_sizes, int n_in,
                              void* d_out, int out_size, void* d_ws, size_t ws_size,
                              hipStream_t stream) {
    const float* x  = (const float*)d_in[0];  // [B, D_INNER, L]
    const float* dA = (const float*)d_in[1];  // [B, D_INNER, L, D_STATE]
    const float* dB = (const float*)d_in[2];  // [B, D_INNER, L, D_STATE]
    const float* C  = (const float*)d_in[3];  // [B, D_STATE, L]
    const float* D  = (const float*)d_in[4];  // [D_INNER]
    float* y = (float*)d_out;                 // [B, D_INNER, L]

    (void)in_sizes; (void)n_in; (void)out_size; (void)d_ws; (void)ws_size;

    dim3 grid(BATCH * BLOCKS_PER_BATCH);      // 384 blocks
    dim3 block(THREADS);                      // 128 threads = 4 wave32 waves
    ssm_scan_kernel<<<grid, block, 0, stream>>>(x, dA, dB, C, D, y);
}